// MIND3D_76158360092738
// MI455X (gfx1250) — compile-verified
//
#include <hip/hip_runtime.h>
#include <stdint.h>

// MIND 3D descriptor, fused single-pass kernel for MI455X (gfx1250).
// Memory-bound (~430MB traffic -> ~18us roofline @ 23.3 TB/s). WMMA is not
// applicable (depthwise separable 3-tap stencils; inflating them into 16x16
// matmuls adds ~5x flops for no roofline gain). CDNA5 levers used instead:
//   * Tensor Data Mover (tensor_load_to_lds + s_wait_tensorcnt) async-stages
//     image planes into LDS (rolling 5-plane window), overlapped with compute.
//     Emission verified in disasm: "tensor_load_to_lds s[0:3], s[4:11]" with
//     the exact D# fields (lds_addr, type=2, data_size=4B, tile 36x12,
//     stride 192) built below.
//   * wave32-shaped 32x8 blocks: every wave stores full 128B cachelines.
//   * non-temporal stores for the 340MB streaming output; input (57MB) is
//     left cacheable since it fits in the 192MB L2 (halo/chunk re-reads hit).

typedef uint32_t v4u __attribute__((ext_vector_type(4)));
typedef int      v8i __attribute__((ext_vector_type(8)));
typedef int      v4i __attribute__((ext_vector_type(4)));

#define IMG_W 192
#define IMG_H 192
#define IMG_D 192
#define NB    2
#define NCH   6
#define TX    32
#define TY    8
#define HALO  2
#define LW    (TX + 2 * HALO)   // 36
#define LH    (TY + 2 * HALO)   // 12
#define PLANE (LW * LH)         // 432
#define NTHR  (TX * TY)         // 256 = 8 waves
#define ZCHUNK 48
#define NZCH  (IMG_D / ZCHUNK)  // 4
#define EPS   1e-8f
#define GW    0.018315638888734179f  // exp(-4): per-unit-offset Gaussian weight

// ---------------------------------------------------------------------------
// TDM: DMA one 36x12 f32 tile (row pitch = IMG_W elements) into LDS.
// Descriptor layout per CDNA5 ISA ch.8 (Tensor DMA Descriptor D#).
// ---------------------------------------------------------------------------
__device__ __forceinline__ void tdm_load_plane_to_lds(const float* gsrc,
                                                      uint32_t lds_byte) {
  const uint64_t ga = (uint64_t)(uintptr_t)gsrc;
  v4u g0;
  g0.x = 1u;                                   // count=1, user mode, no gather
  g0.y = lds_byte;                             // lds_addr (bytes)
  g0.z = (uint32_t)ga;                         // global_addr[31:0]
  g0.w = (uint32_t)((ga >> 32) & 0x01FFFFFFu)  // global_addr[56:32]
         | (2u << 30);                         // type = 2 (image)
  const uint32_t td0 = 1u << 20;               // tensor dims large: interior
  const uint32_t td1 = 1u << 20;               // tiles never go OOB
  v8i g1;
  g1[0] = (int)(2u << 16);                     // wg_mask=0, data_size=2 (4B)
  g1[1] = (int)((td0 & 0xFFFFu) << 16);        // atomic_addr=0, tensor_dim0 lo
  g1[2] = (int)(((td0 >> 16) & 0xFFFFu) | ((td1 & 0xFFFFu) << 16));
  g1[3] = (int)(((td1 >> 16) & 0xFFFFu) | ((uint32_t)LW << 16)); // tile_dim0=36
  g1[4] = (int)LH;                             // tile_dim1=12, tile_dim2=0 (2D)
  g1[5] = (int)IMG_W;                          // tensor_dim0_stride = 192 elems
  g1[6] = 0;                                   // stride0 hi / stride1 lo
  g1[7] = 0;                                   // stride1 hi
  v4i gz = {0, 0, 0, 0};
#if defined(__clang_major__) && (__clang_major__ >= 23)
  v8i gz8 = {0, 0, 0, 0, 0, 0, 0, 0};
  __builtin_amdgcn_tensor_load_to_lds(g0, g1, gz, gz, gz8, 0);
#else
  __builtin_amdgcn_tensor_load_to_lds(g0, g1, gz, gz, 0);
#endif
}

// ---------------------------------------------------------------------------
// In-plane 9-tap (separable [w,1,w] x [w,1,w]) smoothing of the 6 per-channel
// squared shift-differences at plane zz, evaluated at this thread's (gx,gy).
// Pm/P0/Pp are LDS planes zz-1 / zz / zz+1 (36x12, halo 2, zero-filled OOB).
// Validity masks implement the conv's zero padding of diff2 outside the domain.
// Channel order matches the reference: +z, -z, +y, -y, +x, -x.
// ---------------------------------------------------------------------------
__device__ __forceinline__ void compute_E(const float* __restrict__ Pm,
                                          const float* __restrict__ P0,
                                          const float* __restrict__ Pp,
                                          int tx, int ty, int gx, int gy,
                                          float* __restrict__ E) {
  float v[5][5];
#pragma unroll
  for (int j = 0; j < 5; ++j)
#pragma unroll
    for (int i = 0; i < 5; ++i)
      v[j][i] = P0[(ty + j) * LW + (tx + i)];
  float zp[3][3], zm[3][3];
#pragma unroll
  for (int j = 0; j < 3; ++j)
#pragma unroll
    for (int i = 0; i < 3; ++i) {
      zp[j][i] = Pp[(ty + 1 + j) * LW + (tx + 1 + i)];
      zm[j][i] = Pm[(ty + 1 + j) * LW + (tx + 1 + i)];
    }
  const float g1[3] = {GW, 1.0f, GW};
  float my[3], mx[3];
#pragma unroll
  for (int k = 0; k < 3; ++k) {
    const int yy = gy - 1 + k;
    const int xx = gx - 1 + k;
    my[k] = (yy >= 0 && yy < IMG_H) ? 1.0f : 0.0f;
    mx[k] = (xx >= 0 && xx < IMG_W) ? 1.0f : 0.0f;
  }
#pragma unroll
  for (int c = 0; c < NCH; ++c) E[c] = 0.0f;
#pragma unroll
  for (int j = 0; j < 3; ++j)
#pragma unroll
    for (int i = 0; i < 3; ++i) {
      const float ctr = v[j + 1][i + 1];
      const float wt = g1[j] * g1[i] * my[j] * mx[i];
      float d;
      d = zp[j][i] - ctr;        E[0] = fmaf(wt, d * d, E[0]);  // +z
      d = zm[j][i] - ctr;        E[1] = fmaf(wt, d * d, E[1]);  // -z
      d = v[j + 2][i + 1] - ctr; E[2] = fmaf(wt, d * d, E[2]);  // +y
      d = v[j][i + 1] - ctr;     E[3] = fmaf(wt, d * d, E[3]);  // -y
      d = v[j + 1][i + 2] - ctr; E[4] = fmaf(wt, d * d, E[4]);  // +x
      d = v[j + 1][i] - ctr;     E[5] = fmaf(wt, d * d, E[5]);  // -x
    }
}

__global__ __launch_bounds__(NTHR) void mind3d_kernel(
    const float* __restrict__ img, float* __restrict__ out) {
  __shared__ float lds[5][PLANE];  // rolling window of 5 z-planes, 36x12 each

  const int tx = threadIdx.x, ty = threadIdx.y;
  const int tid = ty * TX + tx;
  const int x0 = blockIdx.x * TX, y0 = blockIdx.y * TY;
  const int bb = blockIdx.z / NZCH;            // batch
  const int zb = (blockIdx.z % NZCH) * ZCHUNK; // z-chunk start
  const int ze = zb + ZCHUNK;
  const int gx = x0 + tx, gy = y0 + ty;
  const bool interior = (x0 >= HALO) && (x0 + TX + HALO <= IMG_W) &&
                        (y0 >= HALO) && (y0 + TY + HALO <= IMG_H);

  // Block-uniform source base for plane q: src0 + q * (H*W)
  const float* const src0 =
      img + (((size_t)bb * IMG_D * IMG_H + (y0 - HALO)) * IMG_W + (x0 - HALO));

  auto zero_slot = [&](int slot) {
    float* S = &lds[slot][0];
    for (int idx = tid; idx < PLANE; idx += NTHR) S[idx] = 0.0f;
  };
  auto fetch_plane = [&](int q) {  // q in [0, IMG_D)
    const int slot = q % 5;
    if (interior) {
      if (tid < 32) {  // wave 0 only issues the TDM (EXEC-independent DMA)
        tdm_load_plane_to_lds(src0 + (size_t)q * (IMG_H * IMG_W),
                              (uint32_t)(uintptr_t)(&lds[slot][0]));
      }
    } else {  // boundary tile: guarded loads with zero fill (shift padding)
      float* S = &lds[slot][0];
      const size_t base = (size_t)(bb * IMG_D + q) * IMG_H;
      for (int idx = tid; idx < PLANE; idx += NTHR) {
        const int ly = idx / LW, lx = idx - ly * LW;
        const int yy = y0 - HALO + ly, xx = x0 - HALO + lx;
        float v = 0.0f;
        if (yy >= 0 && yy < IMG_H && xx >= 0 && xx < IMG_W)
          v = img[(base + yy) * IMG_W + xx];
        S[idx] = v;
      }
    }
  };

  // Prologue: stage planes zb-2 .. zb+2 (virtual planes < 0 are zero).
  {
    const int qm2 = zb - 2, qm1 = zb - 1;
    if (qm2 < 0) zero_slot(3); else fetch_plane(qm2);  // plane -2 -> slot 3
    if (qm1 < 0) zero_slot(4); else fetch_plane(qm1);  // plane -1 -> slot 4
    fetch_plane(zb);
    fetch_plane(zb + 1);
    fetch_plane(zb + 2);
  }
  __builtin_amdgcn_s_wait_tensorcnt(0);
  __syncthreads();

  float Em1[NCH], E0[NCH], Ep1[NCH];
  if (zb - 1 >= 0)
    compute_E(&lds[(((zb - 2) % 5) + 5) % 5][0],
              &lds[(((zb - 1) % 5) + 5) % 5][0], &lds[zb % 5][0],
              tx, ty, gx, gy, Em1);
  else
#pragma unroll
    for (int c = 0; c < NCH; ++c) Em1[c] = 0.0f;
  compute_E(&lds[(((zb - 1) % 5) + 5) % 5][0], &lds[zb % 5][0],
            &lds[(zb + 1) % 5][0], tx, ty, gx, gy, E0);

  const float invn =
      1.0f / ((1.0f + 2.0f * GW) * (1.0f + 2.0f * GW) * (1.0f + 2.0f * GW));

  // Per-channel output pointers, bumped by one plane (H*W) per z step.
  float* op[NCH];
  {
    float* ob = out + (((size_t)bb * NCH * IMG_D + zb) * IMG_H + gy) * IMG_W + gx;
#pragma unroll
    for (int c = 0; c < NCH; ++c)
      op[c] = ob + (size_t)c * IMG_D * IMG_H * IMG_W;
  }

  for (int z = zb; z < ze; ++z) {
    // Overlap: kick off DMA of plane z+3 before computing with planes z..z+2.
    const int pf = z + 3;
    if (pf < IMG_D)       fetch_plane(pf);
    else if (pf == IMG_D) zero_slot(pf % 5);  // zero plane D for E(D-1)'s +z

    if (z + 1 < IMG_D)
      compute_E(&lds[z % 5][0], &lds[(z + 1) % 5][0], &lds[(z + 2) % 5][0],
                tx, ty, gx, gy, Ep1);
    else
#pragma unroll
      for (int c = 0; c < NCH; ++c) Ep1[c] = 0.0f;  // conv zero-pad in z

    float Dp[NCH], s = 0.0f;
#pragma unroll
    for (int c = 0; c < NCH; ++c) {
      Dp[c] = (GW * (Em1[c] + Ep1[c]) + E0[c]) * invn;
      s += Dp[c];
    }
    const float vinv = 1.0f / (s * (1.0f / 6.0f) + EPS);
    float num[NCH], ds = 0.0f;
#pragma unroll
    for (int c = 0; c < NCH; ++c) {
      num[c] = __expf(-Dp[c] * vinv);
      ds += num[c];
    }
    const float r = 1.0f / (ds + EPS);
#pragma unroll
    for (int c = 0; c < NCH; ++c) {
      __builtin_nontemporal_store(num[c] * r, op[c]);  // streaming 128B/wave
      op[c] += IMG_H * IMG_W;
    }

    __builtin_amdgcn_s_wait_tensorcnt(0);  // plane z+3 resident (wave 0)
    __syncthreads();                        // publish LDS to all waves

#pragma unroll
    for (int c = 0; c < NCH; ++c) { Em1[c] = E0[c]; E0[c] = Ep1[c]; }
  }
}

extern "C" void kernel_launch(void* const* d_in, const int* in_sizes, int n_in,
                              void* d_out, int out_size, void* d_ws,
                              size_t ws_size, hipStream_t stream) {
  (void)in_sizes; (void)n_in; (void)out_size; (void)d_ws; (void)ws_size;
  const float* img = (const float*)d_in[0];
  float* out = (float*)d_out;
  dim3 block(TX, TY);
  dim3 grid(IMG_W / TX, IMG_H / TY, NB * NZCH);
  mind3d_kernel<<<grid, block, 0, stream>>>(img, out);
}